// RNN_22522808500864
// MI455X (gfx1250) — compile-verified
//
#include <hip/hip_runtime.h>

#define B 64
#define S 512
#define I 1024
#define H 1024
#define NBLK2 64
#define LDS_STRIDE 1028  // 1024 + 4 floats pad -> conflict-free b64 reads

typedef __attribute__((ext_vector_type(2))) float v2f;
typedef __attribute__((ext_vector_type(8))) float v8f;

// ---------------- init: reset barrier counter, load h0 into ping buffer ----
__global__ void rnn_init_kernel(const float* __restrict__ h0,
                                float* __restrict__ hbuf0,
                                unsigned* __restrict__ ctr) {
    int idx = blockIdx.x * blockDim.x + threadIdx.x;
    if (idx < B * H) hbuf0[idx] = h0[idx];
    if (idx == 0) *ctr = 0u;
}

// ---------------- phase 1: xp[s,b,h] = x[b,s,:] . Wih[h,:] + bih + bhh ----
// grid = (H/256, S), block = 256 (8 waves). Wave owns 32(M=b) x 64(N=h):
// 2x4 tiles of 16x16 -> 6 loads per 8 WMMAs per k-step of 4.
__global__ __launch_bounds__(256)
void rnn_xproj_kernel(const float* __restrict__ x,    // [B,S,I]
                      const float* __restrict__ wih,  // [H,I]
                      const float* __restrict__ bih,  // [H]
                      const float* __restrict__ bhh,  // [H]
                      float* __restrict__ xp) {       // [S,B,H]
    const int s     = blockIdx.y;
    const int lane  = threadIdx.x & 31;
    const int w     = threadIdx.x >> 5;
    const int l15   = lane & 15;
    const int hi    = lane >> 4;            // 0: K pair {0,1}, 1: K pair {2,3}
    const int b0    = (w & 1) * 32;         // 2 M-tiles: rows b0..b0+31
    const int hbase = blockIdx.x * 256 + (w >> 1) * 64;

    // A tiles: lane%16 = M row, lane/16 selects K pair; contiguous K pair -> b64
    const float* arow0 = x + ((size_t)(b0 + l15) * S + s) * I + 2 * hi;
    const float* arow1 = x + ((size_t)(b0 + 16 + l15) * S + s) * I + 2 * hi;
    // B tiles: B[k][n] = Wih[hbase+n][k]; lane%16 = N col
    const float* brow  = wih + (size_t)(hbase + l15) * I + 2 * hi;

    v8f acc[2][4] = {{v8f{}, v8f{}, v8f{}, v8f{}}, {v8f{}, v8f{}, v8f{}, v8f{}}};
#pragma unroll 2
    for (int k = 0; k < I; k += 4) {
        v2f a0  = *(const v2f*)(arow0 + k);
        v2f a1  = *(const v2f*)(arow1 + k);
        v2f bm0 = *(const v2f*)(brow + k);
        v2f bm1 = *(const v2f*)(brow + 16 * I + k);
        v2f bm2 = *(const v2f*)(brow + 32 * I + k);
        v2f bm3 = *(const v2f*)(brow + 48 * I + k);
        acc[0][0] = __builtin_amdgcn_wmma_f32_16x16x4_f32(false, a0, false, bm0, (short)0, acc[0][0], false, false);
        acc[0][1] = __builtin_amdgcn_wmma_f32_16x16x4_f32(false, a0, false, bm1, (short)0, acc[0][1], false, false);
        acc[0][2] = __builtin_amdgcn_wmma_f32_16x16x4_f32(false, a0, false, bm2, (short)0, acc[0][2], false, false);
        acc[0][3] = __builtin_amdgcn_wmma_f32_16x16x4_f32(false, a0, false, bm3, (short)0, acc[0][3], false, false);
        acc[1][0] = __builtin_amdgcn_wmma_f32_16x16x4_f32(false, a1, false, bm0, (short)0, acc[1][0], false, false);
        acc[1][1] = __builtin_amdgcn_wmma_f32_16x16x4_f32(false, a1, false, bm1, (short)0, acc[1][1], false, false);
        acc[1][2] = __builtin_amdgcn_wmma_f32_16x16x4_f32(false, a1, false, bm2, (short)0, acc[1][2], false, false);
        acc[1][3] = __builtin_amdgcn_wmma_f32_16x16x4_f32(false, a1, false, bm3, (short)0, acc[1][3], false, false);
    }

    // C/D layout: lane%16 = N, vgpr r -> M = r + 8*(lane/16)
#pragma unroll
    for (int j = 0; j < 4; ++j) {
        const int hcol = hbase + j * 16 + l15;
        const float bias = bih[hcol] + bhh[hcol];
#pragma unroll
        for (int mi = 0; mi < 2; ++mi) {
#pragma unroll
            for (int r = 0; r < 8; ++r) {
                const int brw = b0 + mi * 16 + r + 8 * hi;
                xp[((size_t)s * B + brw) * H + hcol] = acc[mi][j][r] + bias;
            }
        }
    }
}

// ---------------- phase 2: persistent recurrent scan -----------------------
// h_next = relu(xp[s] + h_cur @ Whh^T). 64 blocks x 4 waves = 256 waves
// = 256 16x16 tiles. All 4 waves of a block share one 16-col Whh strip
// -> stage it in LDS once via async-to-LDS (immune to per-step acquire
// invalidation of WGP$). Grid barrier per step.
__global__ __launch_bounds__(128)
void rnn_recur_kernel(const float* __restrict__ whh,  // [H,H]
                      const float* __restrict__ xp,   // [S,B,H]
                      float* __restrict__ hbuf,       // [2,B,H] ping-pong
                      unsigned* ctr,
                      float* __restrict__ out) {      // [B,S,H] then [B,H]
    __shared__ float bs[16 * LDS_STRIDE];             // ~64.25 KB Whh strip

    const int lane = threadIdx.x & 31;
    const int w    = threadIdx.x >> 5;
    const int l15  = lane & 15;
    const int hi   = lane >> 4;
    const int t    = blockIdx.x * 4 + w;   // tile id: m = t&3, n = t>>2 = blockIdx.x
    const int b0   = (t & 3) * 16;
    const int h0   = blockIdx.x * 16;

    // ---- async bulk fill: Whh[h0..h0+15][0..1023] -> LDS (padded rows) ----
    // 16 rows x 256 b128-chunks = 4096 chunks; 128 threads x 32 iterations.
#pragma unroll 4
    for (int i = 0; i < 32; ++i) {
        const int idx  = threadIdx.x + i * 128;
        const int row  = idx >> 8;
        const int col4 = (idx & 255) * 4;
        const unsigned lds_off =
            (unsigned)(size_t)&bs[row * LDS_STRIDE + col4];
        const unsigned long long gaddr =
            (unsigned long long)(const void*)(whh + (size_t)(h0 + row) * H + col4);
        asm volatile("global_load_async_to_lds_b128 %0, %1, off"
                     :: "v"(lds_off), "v"(gaddr) : "memory");
    }
    asm volatile("s_wait_asynccnt 0" ::: "memory");
    __syncthreads();

    float* hlast = out + (size_t)B * S * H;

    for (int s = 0; s < S; ++s) {
        const float* cur = hbuf + (size_t)(s & 1) * B * H;
        float* nxt       = hbuf + (size_t)((s + 1) & 1) * B * H;
        const float* arow = cur + (size_t)(b0 + l15) * H + 2 * hi;
        const float* bls  = &bs[l15 * LDS_STRIDE + 2 * hi];

        v8f acc = {};
#pragma unroll 4
        for (int k = 0; k < H; k += 4) {
            v2f a  = *(const v2f*)(arow + k);   // global b64
            v2f bm = *(const v2f*)(bls + k);    // ds_load_b64, conflict-free
            acc = __builtin_amdgcn_wmma_f32_16x16x4_f32(false, a, false, bm, (short)0, acc, false, false);
        }

        const int hcol = h0 + l15;
#pragma unroll
        for (int r = 0; r < 8; ++r) {
            const int brw = b0 + r + 8 * hi;
            float v = acc[r] + xp[((size_t)s * B + brw) * H + hcol];
            v = v > 0.0f ? v : 0.0f;
            nxt[(size_t)brw * H + hcol]           = v;
            out[((size_t)brw * S + s) * H + hcol] = v;
            if (s == S - 1) hlast[(size_t)brw * H + hcol] = v;
        }

        // ---- device-wide barrier: release stores, count, acquire ----
        __builtin_amdgcn_fence(__ATOMIC_RELEASE, "agent");
        __syncthreads();
        if (threadIdx.x == 0) {
            __hip_atomic_fetch_add(ctr, 1u, __ATOMIC_RELAXED, __HIP_MEMORY_SCOPE_AGENT);
            const unsigned target = (unsigned)NBLK2 * (unsigned)(s + 1);
            while (__hip_atomic_load(ctr, __ATOMIC_RELAXED, __HIP_MEMORY_SCOPE_AGENT) < target) {
                __builtin_amdgcn_s_sleep(1);
            }
        }
        __syncthreads();
        __builtin_amdgcn_fence(__ATOMIC_ACQUIRE, "agent");  // drop stale WGP$ lines
    }
}

extern "C" void kernel_launch(void* const* d_in, const int* in_sizes, int n_in,
                              void* d_out, int out_size, void* d_ws, size_t ws_size,
                              hipStream_t stream) {
    const float* x   = (const float*)d_in[0];  // [B,S,I]
    const float* h0  = (const float*)d_in[1];  // [1,B,H]
    const float* wih = (const float*)d_in[2];  // [H,I]
    const float* whh = (const float*)d_in[3];  // [H,H]
    const float* bih = (const float*)d_in[4];  // [H]
    const float* bhh = (const float*)d_in[5];  // [H]
    float* out = (float*)d_out;                // [B,S,H] ++ [1,B,H]

    float* xp      = (float*)d_ws;                       // S*B*H floats
    float* hbuf    = xp + (size_t)S * B * H;             // 2*B*H floats
    unsigned* ctr  = (unsigned*)(hbuf + 2 * (size_t)B * H);

    rnn_init_kernel<<<dim3((B * H + 255) / 256), 256, 0, stream>>>(h0, hbuf, ctr);
    rnn_xproj_kernel<<<dim3(H / 256, S), 256, 0, stream>>>(x, wih, bih, bhh, xp);
    rnn_recur_kernel<<<dim3(NBLK2), 128, 0, stream>>>(whh, xp, hbuf, ctr, out);
}